// Dist_Loss_77060303225417
// MI455X (gfx1250) — compile-verified
//
#include <hip/hip_runtime.h>
#include <math.h>

typedef __attribute__((ext_vector_type(2))) float v2f;
typedef __attribute__((ext_vector_type(8))) float v8f;

#define NUM_CLASSES 10
#define CPAD 16                 // classes padded to WMMA tile height
#define FDIM 64
#define K1_BLOCKS 256
#define K3_BLOCKS 256
#define THREADS 256
#define WAVES_PER_BLOCK (THREADS / 32)

// workspace layout (floats):
//   [2][K1_BLOCKS][CPAD*FDIM + CPAD]   per-block center partials (sums + counts)
//   [2][CPAD*FDIM + CPAD]              reduced sums + counts
//   [2][K3_BLOCKS][CPAD]               per-block distance partials
#define PART_STRIDE (CPAD * FDIM + CPAD)             // 1040
#define OFF_RED (2 * K1_BLOCKS * PART_STRIDE)
#define OFF_DIST (OFF_RED + 2 * PART_STRIDE)

// ---------------------------------------------------------------------------
// Kernel 1: per-class segment sums via V_WMMA_F32_16X16X4_F32.
// A = one-hot(label) tile (16x4), B = feat tile (4x16). Wave processes 4
// samples per iteration across the 64-wide feature dim (4 WMMAs).
// ---------------------------------------------------------------------------
__global__ void k_center_sums(const float* __restrict__ feat1, const int* __restrict__ lab1,
                              const float* __restrict__ feat2, const int* __restrict__ lab2,
                              float* __restrict__ ws, int N) {
    const int dom = blockIdx.y;
    const float* feat = dom ? feat2 : feat1;
    const int* lab = dom ? lab2 : lab1;

    __shared__ float lsum[CPAD * FDIM];
    __shared__ float lcnt[CPAD];
    for (int i = threadIdx.x; i < CPAD * FDIM; i += THREADS) lsum[i] = 0.f;
    if (threadIdx.x < CPAD) lcnt[threadIdx.x] = 0.f;
    __syncthreads();

    const int lane = threadIdx.x & 31;
    const int half = lane >> 4;       // lane half selects K pair {0,1} vs {2,3}
    const int m16  = lane & 15;       // class row for A, feature col for B
    const int wid  = blockIdx.x * WAVES_PER_BLOCK + (threadIdx.x >> 5);
    const int W    = K1_BLOCKS * WAVES_PER_BLOCK;   // total waves per domain
    const int iters = N / (4 * W);                  // N=262144, W=2048 -> 32

    v8f acc0 = {}, acc1 = {}, acc2 = {}, acc3 = {};
    float cnt = 0.f;

    for (int it = 0; it < iters; ++it) {
        const int s0 = (wid + it * W) * 4;
        const int sA = s0 + 2 * half;
        const int la = lab[sA];
        const int lb = lab[sA + 1];
        v2f a;
        a.x = (la == m16) ? 1.f : 0.f;   // VGPR0 <-> K = 2*half
        a.y = (lb == m16) ? 1.f : 0.f;   // VGPR1 <-> K = 2*half + 1
        cnt += a.x + a.y;

        const float* r0 = feat + (size_t)sA * FDIM;
        const float* r1 = r0 + FDIM;
        v2f b0, b1, b2, b3;
        b0.x = r0[ 0 + m16]; b0.y = r1[ 0 + m16];
        b1.x = r0[16 + m16]; b1.y = r1[16 + m16];
        b2.x = r0[32 + m16]; b2.y = r1[32 + m16];
        b3.x = r0[48 + m16]; b3.y = r1[48 + m16];

        acc0 = __builtin_amdgcn_wmma_f32_16x16x4_f32(false, a, false, b0, (short)0, acc0, false, false);
        acc1 = __builtin_amdgcn_wmma_f32_16x16x4_f32(false, a, false, b1, (short)0, acc1, false, false);
        acc2 = __builtin_amdgcn_wmma_f32_16x16x4_f32(false, a, false, b2, (short)0, acc2, false, false);
        acc3 = __builtin_amdgcn_wmma_f32_16x16x4_f32(false, a, false, b3, (short)0, acc3, false, false);
    }

    // C/D layout: VGPR v holds row M = v + 8*half, col = lane%16
#pragma unroll
    for (int v = 0; v < 8; ++v) {
        const int row = v + 8 * half;
        atomicAdd(&lsum[row * FDIM +  0 + m16], acc0[v]);
        atomicAdd(&lsum[row * FDIM + 16 + m16], acc1[v]);
        atomicAdd(&lsum[row * FDIM + 32 + m16], acc2[v]);
        atomicAdd(&lsum[row * FDIM + 48 + m16], acc3[v]);
    }
    atomicAdd(&lcnt[m16], cnt);
    __syncthreads();

    float* out = ws + ((size_t)dom * K1_BLOCKS + blockIdx.x) * PART_STRIDE;
    for (int i = threadIdx.x; i < CPAD * FDIM; i += THREADS) out[i] = lsum[i];
    if (threadIdx.x < CPAD) out[CPAD * FDIM + threadIdx.x] = lcnt[threadIdx.x];
}

// ---------------------------------------------------------------------------
// Kernel 2: fixed-order reduction of per-block partials -> sums + counts
// ---------------------------------------------------------------------------
__global__ void k_reduce_centers(float* __restrict__ ws) {
    const int dom = blockIdx.x;
    const float* part = ws + (size_t)dom * K1_BLOCKS * PART_STRIDE;
    float* red = ws + OFF_RED + (size_t)dom * PART_STRIDE;
    for (int e = threadIdx.x; e < PART_STRIDE; e += THREADS) {
        float s = 0.f;
        for (int b = 0; b < K1_BLOCKS; ++b) s += part[(size_t)b * PART_STRIDE + e];
        red[e] = s;
    }
}

// ---------------------------------------------------------------------------
// Kernel 3: per-sample distance to own-class center; half-wave per sample
// (16 lanes x float4 = one full 256B feature row, perfectly coalesced).
// ---------------------------------------------------------------------------
__global__ void k_dist(const float* __restrict__ feat1, const int* __restrict__ lab1,
                       const float* __restrict__ feat2, const int* __restrict__ lab2,
                       float* __restrict__ ws, int N) {
    const int dom = blockIdx.y;
    const float* feat = dom ? feat2 : feat1;
    const int* lab = dom ? lab2 : lab1;
    const float* red = ws + OFF_RED + (size_t)dom * PART_STRIDE;

    __shared__ __align__(16) float lcent[CPAD * FDIM];
    __shared__ float ldist[CPAD];
    for (int e = threadIdx.x; e < CPAD * FDIM; e += THREADS) {
        const int c = e >> 6;
        lcent[e] = red[e] / fmaxf(red[CPAD * FDIM + c], 1.f);
    }
    if (threadIdx.x < CPAD) ldist[threadIdx.x] = 0.f;
    __syncthreads();

    const int lane16 = threadIdx.x & 15;
    const int hw = blockIdx.x * (THREADS / 16) + (threadIdx.x >> 4);
    const int HW = K3_BLOCKS * (THREADS / 16);
    const float4* feat4 = (const float4*)feat;
    const float4* cent4 = (const float4*)lcent;

    for (int s = hw; s < N; s += HW) {
        const int lb = lab[s];
        float4 f = feat4[(size_t)s * 16 + lane16];
        float4 c = cent4[lb * 16 + lane16];
        float dx = f.x - c.x, dy = f.y - c.y, dz = f.z - c.z, dw = f.w - c.w;
        float ss = dx * dx + dy * dy + dz * dz + dw * dw;
        ss += __shfl_xor(ss, 1, 32);
        ss += __shfl_xor(ss, 2, 32);
        ss += __shfl_xor(ss, 4, 32);
        ss += __shfl_xor(ss, 8, 32);
        if (lane16 == 0) atomicAdd(&ldist[lb], sqrtf(ss));
    }
    __syncthreads();
    if (threadIdx.x < CPAD)
        ws[OFF_DIST + ((size_t)dom * K3_BLOCKS + blockIdx.x) * CPAD + threadIdx.x] =
            ldist[threadIdx.x];
}

// ---------------------------------------------------------------------------
// Kernel 4: final epilogue (tiny): reduce distance partials, 10x10 pairwise
// center distances, scalar loss.
// ---------------------------------------------------------------------------
__global__ void k_final(const float* __restrict__ ws, float* __restrict__ out) {
    __shared__ float cent[2 * CPAD * FDIM];
    __shared__ float dsum[2 * CPAD];
    __shared__ float pd[128];
    const int tid = threadIdx.x;

    for (int e = tid; e < 2 * CPAD * FDIM; e += blockDim.x) {
        const int dom = e >> 10;
        const int idx = e & 1023;
        const int c = idx >> 6;
        const float* red = ws + OFF_RED + (size_t)dom * PART_STRIDE;
        cent[e] = red[idx] / fmaxf(red[CPAD * FDIM + c], 1.f);
    }
    if (tid < 2 * CPAD) {
        const int dom = tid >> 4;
        const int c = tid & 15;
        const float* dp = ws + OFF_DIST + (size_t)dom * K3_BLOCKS * CPAD;
        float s = 0.f;
        for (int b = 0; b < K3_BLOCKS; ++b) s += dp[b * CPAD + c];
        dsum[tid] = s;
    }
    __syncthreads();

    float v = 0.f;
    if (tid < NUM_CLASSES * NUM_CLASSES) {
        const int i = tid / NUM_CLASSES, j = tid % NUM_CLASSES;
        float ss = 0.f;
        for (int d = 0; d < FDIM; ++d) {
            float df = cent[i * FDIM + d] - cent[CPAD * FDIM + j * FDIM + d];
            ss += df * df;
        }
        const float* red0 = ws + OFF_RED;
        const float* red1 = ws + OFF_RED + PART_STRIDE;
        const bool valid = (red0[CPAD * FDIM + i] > 0.f) && (red1[CPAD * FDIM + j] > 0.f);
        v = valid ? sqrtf(ss) : 0.f;
    }
    if (tid < 128) pd[tid] = v;
    __syncthreads();

    if (tid == 0) {
        const float* red0 = ws + OFF_RED;
        const float* red1 = ws + OFF_RED + PART_STRIDE;
        float intra = 0.f, nval = 0.f, pdsum = 0.f;
        for (int c = 0; c < NUM_CLASSES; ++c) {
            const float c1 = red0[CPAD * FDIM + c];
            const float c2 = red1[CPAD * FDIM + c];
            const float m1 = dsum[c] / fmaxf(c1, 1.f);
            const float m2 = dsum[CPAD + c] / fmaxf(c2, 1.f);
            if (c1 > 1.f && c2 > 1.f) intra += m1 + m2;
            if (c1 > 0.f && c2 > 0.f) nval += 1.f;
        }
        for (int p = 0; p < NUM_CLASSES * NUM_CLASSES; ++p) pdsum += pd[p];
        const float inter = (nval > 1.f) ? pdsum / fmaxf(nval * nval, 1.f) : 0.f;
        const float normalized = intra / (inter + 1e-8f);
        const float loss = (inter > 0.f) ? log1pf(expf(normalized / 10.f)) : intra;
        out[0] = loss;
    }
}

extern "C" void kernel_launch(void* const* d_in, const int* in_sizes, int n_in,
                              void* d_out, int out_size, void* d_ws, size_t ws_size,
                              hipStream_t stream) {
    const float* feat1 = (const float*)d_in[0];
    const int*   lab1  = (const int*)d_in[1];
    const float* feat2 = (const float*)d_in[2];
    const int*   lab2  = (const int*)d_in[3];
    float* out = (float*)d_out;
    float* ws  = (float*)d_ws;
    const int N = in_sizes[1];   // 262144

    k_center_sums<<<dim3(K1_BLOCKS, 2), THREADS, 0, stream>>>(feat1, lab1, feat2, lab2, ws, N);
    k_reduce_centers<<<dim3(2), THREADS, 0, stream>>>(ws);
    k_dist<<<dim3(K3_BLOCKS, 2), THREADS, 0, stream>>>(feat1, lab1, feat2, lab2, ws, N);
    k_final<<<1, 128, 0, stream>>>(ws, out);
}